// SimpleNet_20770461843813
// MI455X (gfx1250) — compile-verified
//
#include <hip/hip_runtime.h>
#include <hip/hip_bf16.h>

// ---------------------------------------------------------------------------
// score = X·F^T  -  (sum |x-f|)/5 ; then log_softmax over F axis.
// X: [1024, 784] f32, F: [512, 784] f32, out: [1024, 512] f32.
//
// GEMM part -> V_WMMA_F32_16X16X4_F32 (exact fp32 matrix op, CDNA5).
// L1 part   -> VALU (sub + abs-add), fed from LDS tiles shared by 8 waves.
// ---------------------------------------------------------------------------

#define D_DIM 784
#define KC 112          // K-chunk (7 * 112 = 784)
#define NCHUNK 7
#define LDS_STRIDE 116  // KC + 4 pad: 16 consecutive rows hit distinct banks
#define BT 32           // B rows per workgroup
#define FT 64           // F cols per workgroup

typedef __attribute__((ext_vector_type(2))) float v2f;
typedef __attribute__((ext_vector_type(8))) float v8f;

__global__ __launch_bounds__(256)
void score_kernel(const float* __restrict__ X, const float* __restrict__ Fm,
                  float* __restrict__ out, int Fn) {
  __shared__ float xs[BT * LDS_STRIDE];  // 32 x 112 tile of X
  __shared__ float fs[FT * LDS_STRIDE];  // 64 x 112 tile of F

  const int tid  = threadIdx.x;
  const int lane = tid & 31;
  const int wave = tid >> 5;   // 8 waves
  const int bi   = wave >> 2;  // 0..1 : which 16-row sub-tile of B
  const int fj   = wave & 3;   // 0..3 : which 16-col sub-tile of F

  const int rowBase = blockIdx.y * BT;
  const int colBase = blockIdx.x * FT;

  const int l    = lane & 15;   // N index within 16x16 tile
  const int hi   = lane >> 4;   // half-wave select
  const int koff = hi * 2;      // A/B fragment K offset per ISA 16x4 f32 layout

  v8f acc = {};                 // WMMA C/D accumulator (16x16 f32, 8 VGPRs)
  float l1[8] = {};             // L1 accumulators, same lane mapping as C

  for (int kc = 0; kc < NCHUNK; ++kc) {
    const int kBase = kc * KC;
    __syncthreads();  // previous chunk fully consumed before overwrite

    // Cooperative stage: X tile (32 rows x 112) as float4
    for (int t = tid; t < BT * (KC / 4); t += 256) {
      const int r = t / (KC / 4);
      const int c = t % (KC / 4);
      const float4 v =
          *(const float4*)(X + (size_t)(rowBase + r) * D_DIM + kBase + c * 4);
      *(float4*)(&xs[r * LDS_STRIDE + c * 4]) = v;
    }
    // Cooperative stage: F tile (64 rows x 112) as float4
    for (int t = tid; t < FT * (KC / 4); t += 256) {
      const int r = t / (KC / 4);
      const int c = t % (KC / 4);
      const float4 v =
          *(const float4*)(Fm + (size_t)(colBase + r) * D_DIM + kBase + c * 4);
      *(float4*)(&fs[r * LDS_STRIDE + c * 4]) = v;
    }
    __syncthreads();

    // ---- similarity: 28 x V_WMMA_F32_16X16X4_F32 per chunk ----
    // A (16x4 f32): lanes 0-15 hold K=0,1 ; lanes 16-31 hold K=2,3 (M = lane&15)
    // B (4x16 f32): mirrored layout, N = lane&15
    const float* xrow = &xs[(bi * 16 + l) * LDS_STRIDE + koff];
    const float* frow = &fs[(fj * 16 + l) * LDS_STRIDE + koff];
#pragma unroll 4
    for (int k = 0; k < KC; k += 4) {
      v2f a = *(const v2f*)(xrow + k);
      v2f b = *(const v2f*)(frow + k);
      acc = __builtin_amdgcn_wmma_f32_16x16x4_f32(
          /*neg_a=*/false, a, /*neg_b=*/false, b,
          /*c_mod=*/(short)0, acc, /*reuse_a=*/false, /*reuse_b=*/false);
    }

    // ---- L1 distance: VALU, mapped onto the same C-matrix lane layout ----
    // lane handles col j = fj*16 + l, rows M = hi*8 .. hi*8+7
    const float* fL = &fs[(fj * 16 + l) * LDS_STRIDE];
    const float* xL = &xs[(bi * 16 + hi * 8) * LDS_STRIDE];
    for (int d = 0; d < KC; d += 4) {
      const float4 fv = *(const float4*)(fL + d);
#pragma unroll
      for (int m = 0; m < 8; ++m) {
        const float4 xv = *(const float4*)(xL + m * LDS_STRIDE + d);
        l1[m] += fabsf(xv.x - fv.x) + fabsf(xv.y - fv.y) +
                 fabsf(xv.z - fv.z) + fabsf(xv.w - fv.w);
      }
    }
  }

  // score = similarity - L1/5 ; C layout: VGPR v -> M = v + hi*8, N = l
  const int col = colBase + fj * 16 + l;
#pragma unroll
  for (int v = 0; v < 8; ++v) {
    const int row = rowBase + bi * 16 + hi * 8 + v;
    out[(size_t)row * Fn + col] = acc[v] - l1[v] * 0.2f;
  }
}

// In-place row-wise log_softmax over Fn = 512 columns. One block per row.
__global__ __launch_bounds__(512)
void lsm_kernel(float* __restrict__ out, int Fn) {
  __shared__ float buf[512];
  const int tid = threadIdx.x;
  float* row = out + (size_t)blockIdx.x * Fn;
  const float s = row[tid];

  buf[tid] = s;
  __syncthreads();
  for (int off = 256; off > 0; off >>= 1) {
    if (tid < off) buf[tid] = fmaxf(buf[tid], buf[tid + off]);
    __syncthreads();
  }
  const float mx = buf[0];
  __syncthreads();

  buf[tid] = __expf(s - mx);
  __syncthreads();
  for (int off = 256; off > 0; off >>= 1) {
    if (tid < off) buf[tid] += buf[tid + off];
    __syncthreads();
  }
  const float lse = __logf(buf[0]) + mx;

  row[tid] = s - lse;
}

extern "C" void kernel_launch(void* const* d_in, const int* in_sizes, int n_in,
                              void* d_out, int out_size, void* d_ws, size_t ws_size,
                              hipStream_t stream) {
  const float* X  = (const float*)d_in[0];  // [B, 1, 28, 28] -> [B, 784]
  const float* Fm = (const float*)d_in[1];  // [F, 1, 28, 28] -> [F, 784]
  float* out = (float*)d_out;               // [B, F]

  const int Bn = in_sizes[0] / D_DIM;  // 1024
  const int Fn = in_sizes[1] / D_DIM;  // 512

  dim3 grid(Fn / FT, Bn / BT);  // (8, 32)
  score_kernel<<<grid, 256, 0, stream>>>(X, Fm, out, Fn);
  lsm_kernel<<<Bn, Fn, 0, stream>>>(out, Fn);
}